// ProposalLayer_49864570306634
// MI455X (gfx1250) — compile-verified
//
#include <hip/hip_runtime.h>
#include <stdint.h>

#define BATCH   4
#define NBOX    100000
#define PRE     6000
#define POST    1000
#define SORTN   8192
#define BLOCKS  188      // ceil(PRE/32)
#define KWORDS  192      // padded stride for keep mask
#define IMG_WH  800.0f
#define MINSZ   16.0f
#define NMS_TH  0.7f

// ---- workspace layout (bytes) ----
#define OFF_KEYS  ((size_t)0)                      // BATCH*NBOX u32      = 1,600,000
#define OFF_TBOX  ((size_t)1600000)                // BATCH*PRE float4    =   384,000
#define OFF_TSC   ((size_t)1984000)                // BATCH*PRE float     =    96,000
#define OFF_KEEP  ((size_t)2080000)                // BATCH*KWORDS u32    =     3,072
#define OFF_MASK  ((size_t)2083072)                // BATCH*PRE*BLOCKS u32= 18,048,000

// ---- CDNA5 async global->LDS staging (guarded) ----
#if defined(__has_builtin)
#if __has_builtin(__builtin_amdgcn_global_load_async_to_lds_b128)
#define HAVE_ASYNC_LDS 1
#endif
#endif

#define AS1 __attribute__((address_space(1)))
#define AS3 __attribute__((address_space(3)))

typedef __attribute__((ext_vector_type(4))) int v4i;
typedef AS1 v4i gv4i;   // global int4 (parameter type per compiler diagnostic)
typedef AS3 v4i lv4i;   // LDS int4

__device__ __forceinline__ void async_copy16(const void* g, void* l) {
#if defined(HAVE_ASYNC_LDS)
    __builtin_amdgcn_global_load_async_to_lds_b128(
        (gv4i*)(uintptr_t)g,
        (lv4i*)(unsigned int)(uintptr_t)l, 0, 0);
#else
    *(float4*)l = *(const float4*)g;
#endif
}

__device__ __forceinline__ void wait_async() {
#if defined(HAVE_ASYNC_LDS)
#if __has_builtin(__builtin_amdgcn_s_wait_asynccnt)
    __builtin_amdgcn_s_wait_asynccnt(0);
#else
    asm volatile("s_wait_asynccnt 0" ::: "memory");
#endif
#endif
}

__device__ __forceinline__ float clipf(float v) {
    return fminf(fmaxf(v, 0.0f), IMG_WH);
}

// ================= kernel 1: sort keys =================
__global__ __launch_bounds__(256) void k_keys(const float4* __restrict__ props,
                                              const float*  __restrict__ cls,
                                              unsigned*     __restrict__ keys) {
    int i = blockIdx.x * blockDim.x + threadIdx.x;
    if (i >= BATCH * NBOX) return;
    float4 p = props[i];
    float x1 = clipf(p.x), y1 = clipf(p.y), x2 = clipf(p.z), y2 = clipf(p.w);
    bool valid = (x2 - x1 >= MINSZ) && (y2 - y1 >= MINSZ);
    float s = cls[(size_t)i * 2 + 1];
    float ms = valid ? s : -__builtin_inff();
    unsigned u = __float_as_uint(ms);
    keys[i] = (u & 0x80000000u) ? ~u : (u | 0x80000000u);
}

// ============ kernel 2: radix-select top-6000 + bitonic sort + gather ============
__global__ __launch_bounds__(1024) void k_select(const float4*  __restrict__ props,
                                                 const unsigned* __restrict__ keys,
                                                 float4* __restrict__ top_boxes,
                                                 float*  __restrict__ top_scores) {
    __shared__ unsigned long long s_kv[SORTN];     // 64 KB
    unsigned* s_tail = (unsigned*)(s_kv + PRE);    // scratch beyond the 6000 kv slots
    unsigned* s_hist = s_tail;                     // 256
    unsigned* s_wA   = s_tail + 256;               // 32
    unsigned* s_wB   = s_tail + 288;               // 32
    unsigned* s_sc   = s_tail + 320;               // scalars

    int tid = threadIdx.x, b = blockIdx.x;
    int lane = tid & 31, wave = tid >> 5;
    const unsigned* kb = keys + (size_t)b * NBOX;

    // ---- stage A: MSB radix find rank-PRE threshold key ----
    if (tid == 0) { s_sc[0] = 0; s_sc[1] = PRE; }
    __syncthreads();
    for (int p = 3; p >= 0; --p) {
        for (int h = tid; h < 256; h += 1024) s_hist[h] = 0;
        __syncthreads();
        unsigned pfx = s_sc[0];
        int shift = 8 * p;
        for (int i = tid; i < NBOX; i += 1024) {
            unsigned k = kb[i];
            if (((k >> shift) >> 8) == pfx)        // prefix match (==0 for p==3)
                atomicAdd(&s_hist[(k >> shift) & 0xFFu], 1u);
        }
        __syncthreads();
        if (tid == 0) {
            unsigned r = s_sc[1], cum = 0;
            int v = 255;
            for (; v > 0; --v) {
                unsigned c = s_hist[v];
                if (cum + c >= r) break;
                cum += c;
            }
            s_sc[1] = r - cum;
            s_sc[0] = (pfx << 8) | (unsigned)v;
        }
        __syncthreads();
    }
    unsigned thr   = s_sc[0];
    unsigned Etake = s_sc[1];       // #equal-to-thr entries to take (smallest index)
    unsigned G     = PRE - Etake;   // #strictly-greater entries

    // ---- stage B: stable ballot-scan compaction into LDS ----
    if (tid == 0) { s_sc[5] = 0; s_sc[6] = 0; }
    __syncthreads();
    for (int base = 0; base < NBOX; base += 1024) {
        int i = base + tid;
        unsigned k = (i < NBOX) ? kb[i] : 0u;
        bool pg = (i < NBOX) && (k > thr);
        bool pe = (i < NBOX) && (k == thr);
        unsigned mg = (unsigned)__ballot(pg);
        unsigned me = (unsigned)__ballot(pe);
        if (lane == 0) { s_wA[wave] = __popc(mg); s_wB[wave] = __popc(me); }
        __syncthreads();
        if (tid == 0) {
            unsigned ag = s_sc[5], ae = s_sc[6];
            for (int w = 0; w < 32; ++w) {
                unsigned tg = s_wA[w], te = s_wB[w];
                s_wA[w] = ag; s_wB[w] = ae;
                ag += tg; ae += te;
            }
            s_sc[5] = ag; s_sc[6] = ae;
        }
        __syncthreads();
        unsigned lt = (1u << lane) - 1u;
        if (pg) {
            unsigned pos = s_wA[wave] + __popc(mg & lt);
            s_kv[pos] = ((unsigned long long)k << 32) | (unsigned)(0xFFFFFFFFu - (unsigned)i);
        }
        if (pe) {
            unsigned epos = s_wB[wave] + __popc(me & lt);
            if (epos < Etake)
                s_kv[G + epos] = ((unsigned long long)k << 32) | (unsigned)(0xFFFFFFFFu - (unsigned)i);
        }
        __syncthreads();
    }
    // pad (overwrites scratch — scalars already in registers)
    for (int i = PRE + tid; i < SORTN; i += 1024) s_kv[i] = 0ull;
    __syncthreads();

    // ---- stage C: bitonic sort descending (ties -> smaller index first via ~idx) ----
    for (int k2 = 2; k2 <= SORTN; k2 <<= 1) {
        for (int j = k2 >> 1; j > 0; j >>= 1) {
            for (int i = tid; i < SORTN; i += 1024) {
                int l = i ^ j;
                if (l > i) {
                    unsigned long long a = s_kv[i], c = s_kv[l];
                    bool desc = ((i & k2) == 0);
                    if (desc ? (a < c) : (a > c)) { s_kv[i] = c; s_kv[l] = a; }
                }
            }
            __syncthreads();
        }
    }

    // ---- stage D: gather clipped boxes + masked scores for sorted top-6000 ----
    const float4* pb = props + (size_t)b * NBOX;
    for (int t = tid; t < PRE; t += 1024) {
        unsigned long long kv = s_kv[t];
        unsigned idx = 0xFFFFFFFFu - (unsigned)(kv & 0xFFFFFFFFull);
        unsigned key = (unsigned)(kv >> 32);
        unsigned u = (key & 0x80000000u) ? (key & 0x7FFFFFFFu) : ~key;
        float sc = __uint_as_float(u);
        float4 p = pb[idx];
        float4 bx = make_float4(clipf(p.x), clipf(p.y), clipf(p.z), clipf(p.w));
        top_boxes[(size_t)b * PRE + t] = bx;
        top_scores[(size_t)b * PRE + t] = sc;
    }
}

// ============ kernel 3: upper-triangle IoU bitmask (async LDS staging) ============
__global__ __launch_bounds__(256) void k_iou(const float4* __restrict__ top_boxes,
                                             unsigned* __restrict__ mask) {
    __shared__ float4 s_cb[1024];
    __shared__ float  s_ca[1024];
    __shared__ float4 s_rb[32];
    __shared__ float  s_ra[32];
    int b  = blockIdx.z;
    int r0 = blockIdx.y * 32;
    int c0 = blockIdx.x * 1024;
    int tid = threadIdx.x;
    const float4* tb = top_boxes + (size_t)b * PRE;

    for (int t = tid; t < 1024; t += 256) {
        int c = c0 + t;
        if (c < PRE) async_copy16(&tb[c], &s_cb[t]);
        else         s_cb[t] = make_float4(0.f, 0.f, 0.f, 0.f);
    }
    if (tid < 32) {
        int r = r0 + tid;
        s_rb[tid] = (r < PRE) ? tb[r] : make_float4(0.f, 0.f, 0.f, 0.f);
    }
    wait_async();
    __syncthreads();
    for (int t = tid; t < 1024; t += 256) {
        float4 v = s_cb[t];
        s_ca[t] = fmaxf(v.z - v.x, 0.f) * fmaxf(v.w - v.y, 0.f);
    }
    if (tid < 32) {
        float4 v = s_rb[tid];
        s_ra[tid] = fmaxf(v.z - v.x, 0.f) * fmaxf(v.w - v.y, 0.f);
    }
    __syncthreads();

    for (int p = tid; p < 1024; p += 256) {        // 32 rows x 32 col-words
        int rr = p >> 5, ww = p & 31;
        int i = r0 + rr;
        if (i >= PRE) continue;
        float4 rb = s_rb[rr];
        float  ra = s_ra[rr];
        unsigned bits = 0;
        int cb0 = ww * 32;
        #pragma unroll 4
        for (int q = 0; q < 32; ++q) {
            int c = c0 + cb0 + q;
            if (c >= PRE) break;
            if (c <= i) continue;
            float4 cb = s_cb[cb0 + q];
            float xx1 = fmaxf(rb.x, cb.x), yy1 = fmaxf(rb.y, cb.y);
            float xx2 = fminf(rb.z, cb.z), yy2 = fminf(rb.w, cb.w);
            float inter = fmaxf(xx2 - xx1, 0.f) * fmaxf(yy2 - yy1, 0.f);
            float uni = ra + s_ca[cb0 + q] - inter;
            if (inter > NMS_TH * fmaxf(uni, 1e-9f)) bits |= (1u << q);
        }
        int gw = blockIdx.x * 32 + ww;
        if (gw < BLOCKS)
            mask[((size_t)b * PRE + i) * BLOCKS + gw] = bits;
    }
}

// ============ kernel 4: serial greedy scan — one wave/image, mask in registers ============
__global__ __launch_bounds__(32) void k_nms_scan(const float* __restrict__ top_scores,
                                                 const unsigned* __restrict__ mask,
                                                 unsigned* __restrict__ keep_out) {
    int b = blockIdx.x, lane = threadIdx.x;
    const float* ts = top_scores + (size_t)b * PRE;

    unsigned kw[6];
    #pragma unroll
    for (int s = 0; s < 6; ++s) {
        int w = lane + 32 * s;
        unsigned bits = 0;
        if (w < BLOCKS) {
            for (int q = 0; q < 32; ++q) {
                int j = w * 32 + q;
                if (j < PRE && ts[j] > -3.0e38f) bits |= (1u << q);  // valid (finite score)
            }
        }
        kw[s] = bits;
    }
    unsigned k0 = kw[0], k1 = kw[1], k2 = kw[2], k3 = kw[3], k4 = kw[4], k5 = kw[5];

    const unsigned* mb = mask + (size_t)b * PRE * BLOCKS;
    const unsigned* r0p = mb;
    unsigned c0 = r0p[lane], c1 = r0p[lane + 32], c2 = r0p[lane + 64];
    unsigned c3 = r0p[lane + 96], c4 = r0p[lane + 128];
    unsigned c5 = (lane + 160 < BLOCKS) ? r0p[lane + 160] : 0u;

    for (int i = 0; i < PRE; ++i) {
        unsigned n0 = 0, n1 = 0, n2 = 0, n3 = 0, n4 = 0, n5 = 0;
        if (i + 1 < PRE) {                          // unconditional row prefetch
            const unsigned* nr = mb + (size_t)(i + 1) * BLOCKS;
            n0 = nr[lane];       n1 = nr[lane + 32]; n2 = nr[lane + 64];
            n3 = nr[lane + 96];  n4 = nr[lane + 128];
            n5 = (lane + 160 < BLOCKS) ? nr[lane + 160] : 0u;
        }
        int w = i >> 5, s = w >> 5, ol = w & 31;
        unsigned src = (s == 0) ? k0 : (s == 1) ? k1 : (s == 2) ? k2
                     : (s == 3) ? k3 : (s == 4) ? k4 : k5;
        unsigned wv = __shfl(src, ol, 32);          // wave-uniform keep bit
        if ((wv >> (i & 31)) & 1u) {
            k0 &= ~c0; k1 &= ~c1; k2 &= ~c2; k3 &= ~c3; k4 &= ~c4; k5 &= ~c5;
        }
        c0 = n0; c1 = n1; c2 = n2; c3 = n3; c4 = n4; c5 = n5;
    }
    unsigned* ko = keep_out + (size_t)b * KWORDS;
    ko[lane]       = k0; ko[lane + 32]  = k1; ko[lane + 64]  = k2;
    ko[lane + 96]  = k3; ko[lane + 128] = k4;
    if (lane + 160 < BLOCKS) ko[lane + 160] = k5;
}

// ============ kernel 5: stable compaction of kept entries -> outputs ============
__global__ __launch_bounds__(1024) void k_output(const float4* __restrict__ top_boxes,
                                                 const float*  __restrict__ top_scores,
                                                 const unsigned* __restrict__ keep,
                                                 float*  __restrict__ out_scores,
                                                 float4* __restrict__ out_boxes) {
    __shared__ unsigned s_w[32];
    __shared__ unsigned s_base;
    int b = blockIdx.x, tid = threadIdx.x;
    int lane = tid & 31, wave = tid >> 5;
    if (tid == 0) s_base = 0;
    __syncthreads();
    const unsigned* kb = keep + (size_t)b * KWORDS;

    for (int base = 0; base < PRE; base += 1024) {
        int t = base + tid;
        bool kp = (t < PRE) && ((kb[t >> 5] >> (t & 31)) & 1u);
        unsigned m = (unsigned)__ballot(kp);
        if (lane == 0) s_w[wave] = __popc(m);
        __syncthreads();
        if (tid == 0) {
            unsigned a = s_base;
            for (int w = 0; w < 32; ++w) { unsigned tw = s_w[w]; s_w[w] = a; a += tw; }
            s_base = a;
        }
        __syncthreads();
        if (kp) {
            unsigned lt = (1u << lane) - 1u;
            unsigned pos = s_w[wave] + __popc(m & lt);
            if (pos < POST) {
                out_scores[(size_t)b * POST + pos] = top_scores[(size_t)b * PRE + t];
                out_boxes [(size_t)b * POST + pos] = top_boxes [(size_t)b * PRE + t];
            }
        }
        __syncthreads();
    }
    unsigned K = s_base;
    int start = (int)(K < POST ? K : POST);
    for (int p = start + tid; p < POST; p += 1024) {
        out_scores[(size_t)b * POST + p] = 0.f;
        out_boxes [(size_t)b * POST + p] = make_float4(0.f, 0.f, 0.f, 0.f);
    }
}

// ============================ launcher ============================
extern "C" void kernel_launch(void* const* d_in, const int* in_sizes, int n_in,
                              void* d_out, int out_size, void* d_ws, size_t ws_size,
                              hipStream_t stream) {
    (void)in_sizes; (void)n_in; (void)out_size; (void)ws_size;
    const float4* props = (const float4*)d_in[0];
    const float*  cls   = (const float*)d_in[1];
    char* ws = (char*)d_ws;

    unsigned* keys      = (unsigned*)(ws + OFF_KEYS);
    float4*   top_boxes = (float4*)  (ws + OFF_TBOX);
    float*    top_sc    = (float*)   (ws + OFF_TSC);
    unsigned* keep      = (unsigned*)(ws + OFF_KEEP);
    unsigned* mask      = (unsigned*)(ws + OFF_MASK);

    float*  out_scores = (float*)d_out;                       // BATCH*POST
    float4* out_boxes  = (float4*)((float*)d_out + BATCH * POST);

    int total = BATCH * NBOX;
    k_keys<<<(total + 255) / 256, 256, 0, stream>>>((const float4*)props, cls, keys);
    k_select<<<BATCH, 1024, 0, stream>>>(props, keys, top_boxes, top_sc);
    dim3 giou((PRE + 1023) / 1024, (PRE + 31) / 32, BATCH);   // 6 x 188 x 4
    k_iou<<<giou, 256, 0, stream>>>(top_boxes, mask);
    k_nms_scan<<<BATCH, 32, 0, stream>>>(top_sc, mask, keep);
    k_output<<<BATCH, 1024, 0, stream>>>(top_boxes, top_sc, keep, out_scores, out_boxes);
}